// ASModel_16896401343306
// MI455X (gfx1250) — compile-verified
//
#include <hip/hip_runtime.h>
#include <hip/hip_bf16.h>
#include <math.h>

typedef __attribute__((ext_vector_type(2))) float v2f;
typedef __attribute__((ext_vector_type(8))) float v8f;

#define HIDDEN      1024
#define SINGLE_DIM  512
#define TREE_H      8
#define BATCH       2048
#define N_NEG       8
#define SCR         6.28318530717958647692f   /* 2*pi */
#define N_ROWS      72    /* 64 neg + 8 p2 (p1 cancels algebraically) */
#define N_TILES     32    /* 512 channels / 16 per WMMA tile */
#define NWAVES      8     /* 256 threads, wave32 */

// Stage half a tile (36 rows x {lo,hi} 64B chunks = 4.5KB) into LDS with
// CDNA5 async copies. 9 x global_load_async_to_lds_b128: each lane moves 16B,
// 4 lanes cover one contiguous 64B row-chunk, 8 row-half chunks per op.
__device__ __forceinline__ void stage_half(const float* __restrict__ emb,
                                           const int* __restrict__ rows,
                                           unsigned lds_base, int row0,
                                           int cbase, int lane)
{
    const int sub = lane & 3;     // 16B piece within a 64B chunk
    const int rh0 = lane >> 2;    // row-half slot 0..7 within a group
    #pragma unroll
    for (int g = 0; g < 9; ++g) {
        const int rhl = g * 8 + rh0;             // local row-half 0..71
        const int row = rows[row0 + (rhl >> 1)];
        const int hi  = rhl & 1;                 // 0 -> c, 1 -> c+512
        const unsigned goff = (unsigned)row * (HIDDEN * 4u)
                            + (unsigned)hi  * (SINGLE_DIM * 4u)
                            + (unsigned)cbase * 4u
                            + (unsigned)sub * 16u;
        const unsigned ldsa = lds_base + (unsigned)rhl * 64u + (unsigned)sub * 16u;
        asm volatile("global_load_async_to_lds_b128 %0, %1, %2"
                     :: "v"(ldsa), "v"(goff), "s"(emb)
                     : "memory");
    }
}

// One workgroup per batch element; each of the 8 waves owns 4 channel tiles.
// Per tile: D(16x16) = A(16x72) x B(72x16) via 18 accumulating
// V_WMMA_F32_16X16X4_F32 (A = constant one-hot group selector: rows 0..7 pick
// neg n's 8 tree levels, row 8 picks p2). B holds e(row,c)=emb[c]+emb[c+512]
// built from async-staged LDS chunks.
__global__ __launch_bounds__(256) void punish_kernel(
    const float* __restrict__ emb,        // (N_NODES, 1024) f32
    const int*   __restrict__ pos_path,   // (B, 2, 8) i32
    const int*   __restrict__ neg_path,   // (B, 8, 8) i32
    float*       __restrict__ partial)    // (B) f32
{
    __shared__ int   s_rows[N_ROWS];                 // [0..63]=neg, [64..71]=p2
    __shared__ float s_k[N_NEG];                     // diff_neg_raw - 1
    __shared__ float s_dneg[N_NEG];                  // max(diff_neg_raw, 1)
    __shared__ float s_dpos;                         // max(8 - inter(p1,p2), 1)
    __shared__ float s_wave_sq[NWAVES][N_NEG];
    __shared__ float s_buf[NWAVES][2][72 * 16];      // 2 half-tile buffers/wave (72KB)

    const int b    = blockIdx.x;
    const int tid  = threadIdx.x;
    const int lane = tid & 31;
    const int wave = tid >> 5;
    const int h2   = lane >> 4;   // which half of the wave (WMMA K pairing)
    const int lc   = lane & 15;   // M for A rows, N (channel) for B/D columns

    const int* p1 = pos_path + (size_t)b * 2 * TREE_H;
    const int* p2 = p1 + TREE_H;
    const int* ng = neg_path + (size_t)b * N_NEG * TREE_H;

    if (tid < 64)       s_rows[tid] = ng[tid];
    else if (tid < 72)  s_rows[tid] = p2[tid - 64];

    // Tree-path intersection counts (tiny scalar side computation).
    if (tid < N_NEG) {
        int cnt = 0;
        for (int i = 0; i < TREE_H; ++i) {
            int v = p1[i]; int any = 0;
            for (int j = 0; j < TREE_H; ++j) any |= (v == ng[tid * TREE_H + j]);
            cnt += any;
        }
        float raw   = (float)(TREE_H - cnt);
        s_k[tid]    = raw - 1.0f;
        s_dneg[tid] = fmaxf(raw, 1.0f);
    } else if (tid == N_NEG) {
        int cnt = 0;
        for (int i = 0; i < TREE_H; ++i) {
            int v = p1[i]; int any = 0;
            for (int j = 0; j < TREE_H; ++j) any |= (v == p2[j]);
            cnt += any;
        }
        s_dpos = fmaxf((float)(TREE_H - cnt), 1.0f);
    }
    __syncthreads();

    const unsigned buf_lds0 = (unsigned)(uintptr_t)&s_buf[wave][0][0];
    const unsigned buf_lds1 = (unsigned)(uintptr_t)&s_buf[wave][1][0];

    float sq[N_NEG];
    #pragma unroll
    for (int n = 0; n < N_NEG; ++n) sq[n] = 0.0f;

    v8f acc = {};

    // Prologue: stage s=0 (tile=wave, rows 0..35).
    stage_half(emb, s_rows, buf_lds0, 0, wave * 16, lane);

    // 8 stages per wave: (tile, half) = (wave + 8*(s>>1), s&1). Uniform per
    // wave -> EXEC all-1s at every WMMA.
    for (int s = 0; s < 8; ++s) {
        const int hidx = s & 1;

        if (hidx == 0) {
            #pragma unroll
            for (int n = 0; n < 8; ++n) acc[n] = 0.0f;
        }

        if (s < 7) {
            // Close LDS WAR window: async LDS writes are unordered vs DS reads.
            asm volatile("s_wait_dscnt 0x0" ::: "memory");
            const int ns = s + 1;
            stage_half(emb, s_rows, (ns & 1) ? buf_lds1 : buf_lds0,
                       (ns & 1) * 36, (wave + 8 * (ns >> 1)) * 16, lane);
            asm volatile("s_wait_asynccnt 0x9" ::: "memory");   // stage s landed
        } else {
            asm volatile("s_wait_asynccnt 0x0" ::: "memory");
        }

        const float* buf = &s_buf[wave][hidx][0];
        #pragma unroll
        for (int jl = 0; jl < 9; ++jl) {
            const int j  = hidx * 9 + jl;     // global K-chunk 0..17
            const int r0 = 4 * jl + 2 * h2;   // local row in this half-stage
            const int r1 = r0 + 1;
            v2f bm;                            // B(4x16): lane = K{2*h2,1+2*h2} @ N=lc
            bm.x = buf[(2 * r0    ) * 16 + lc] + buf[(2 * r0 + 1) * 16 + lc];
            bm.y = buf[(2 * r1    ) * 16 + lc] + buf[(2 * r1 + 1) * 16 + lc];
            const int owner = (j < 16) ? (j >> 1) : 8;   // one-hot selector row
            const float av = (lc == owner) ? 1.0f : 0.0f;
            v2f am; am.x = av; am.y = av;
            acc = __builtin_amdgcn_wmma_f32_16x16x4_f32(
                      false, am, false, bm, (short)0, acc, false, false);
        }

        if (hidx == 1) {
            // D: lane holds N=lc; VGPR r holds M = r + 8*h2.
            // Lower half: acc[n]=S_neg[n]; upper half: acc[0]=S_p2 (M=8).
            float a0 = acc[0];
            float Sp2 = __shfl_xor(a0, 16, 32);
            if (h2 == 0) {
                #pragma unroll
                for (int n = 0; n < N_NEG; ++n) {
                    const float coef = (s_k[n] + s_dpos - s_dneg[n]) * SCR;
                    float tmp = coef + 0.5f * (acc[n] - Sp2);
                    tmp = fmaxf(tmp, 0.0f);
                    sq[n] += tmp * tmp;
                }
            }
        }
    }

    // Wave-level channel reduction (upper-half lanes carry zeros).
    #pragma unroll
    for (int n = 0; n < N_NEG; ++n) {
        float v = sq[n];
        for (int off = 1; off < 32; off <<= 1) v += __shfl_xor(v, off, 32);
        if (lane == 0) s_wave_sq[wave][n] = v;
    }
    __syncthreads();

    if (tid == 0) {
        float p = 0.0f;
        #pragma unroll
        for (int n = 0; n < N_NEG; ++n) {
            float ssum = 0.0f;
            for (int w = 0; w < NWAVES; ++w) ssum += s_wave_sq[w][n];
            p += sqrtf(ssum);
        }
        partial[b] = p;
    }
}

// Deterministic final reduction (fixed tree order, no float atomics).
__global__ __launch_bounds__(256) void reduce_kernel(
    const float* __restrict__ partial, float* __restrict__ out)
{
    __shared__ float s[256];
    float v = 0.0f;
    for (int i = threadIdx.x; i < BATCH; i += 256) v += partial[i];
    s[threadIdx.x] = v;
    __syncthreads();
    for (int st = 128; st > 0; st >>= 1) {
        if (threadIdx.x < st) s[threadIdx.x] += s[threadIdx.x + st];
        __syncthreads();
    }
    if (threadIdx.x == 0) out[0] = s[0];
}

extern "C" void kernel_launch(void* const* d_in, const int* in_sizes, int n_in,
                              void* d_out, int out_size, void* d_ws, size_t ws_size,
                              hipStream_t stream) {
    const float* emb = (const float*)d_in[0];   // node_embedding (200000,1024) f32
    const int*   pos = (const int*)  d_in[1];   // pos_path (2048,2,8) i32
    const int*   neg = (const int*)  d_in[2];   // neg_path (2048,8,8) i32
    float* partial = (float*)d_ws;              // 2048 floats of scratch
    float* out     = (float*)d_out;             // 1 float

    punish_kernel<<<BATCH, 256, 0, stream>>>(emb, pos, neg, partial);
    reduce_kernel<<<1, 256, 0, stream>>>(partial, out);
}